// ScaledDotProductAttention_71141838291869
// MI455X (gfx1250) — compile-verified
//
#include <hip/hip_runtime.h>
#include <hip/hip_bf16.h>

// Fused scaled-dot-product attention for MI455X (gfx1250, wave32).
//   B=16, T=2048, D=64, fp32. Outputs: out[B,T,D] then attn[B,T,T] (concat).
// One workgroup per (batch, 16-row query tile). The 16x2048 score tile lives
// entirely in LDS (CDNA5: 320KB/WGP). QK^T and PV use V_WMMA_F32_16X16X4_F32
// (full fp32 precision; kernel is HBM-bound on the attn write / bias read).

#define B_ 16
#define T_ 2048
#define D_ 64
#define SROW 2052              // padded LDS row stride (floats) -> bank spread
#define SCALE 0.125f           // 1/sqrt(64)

typedef __attribute__((ext_vector_type(2))) float v2f;
typedef __attribute__((ext_vector_type(4))) float v4f;
typedef __attribute__((ext_vector_type(8))) float v8f;

// LDS layout (floats):
//   S    : 16 * SROW        = 32832   (score / prob tile)
//   PART : 8  * 16 * 64     = 8192    (per-wave PV partials)
//   RED  : 256                        (row-segment reduction scratch)
//   RMAX : 16
//   RINV : 16
#define SMEM_FLOATS (16 * SROW + 8 * 16 * 64 + 256 + 16 + 16)
#define SMEM_BYTES  (SMEM_FLOATS * 4)

__global__ __launch_bounds__(256)
void sdpa_fused_kernel(const float* __restrict__ q,
                       const float* __restrict__ k,
                       const float* __restrict__ v,
                       const float* __restrict__ bias,
                       float* __restrict__ out,
                       float* __restrict__ attn) {
    extern __shared__ float smem[];
    float* S    = smem;
    float* PART = S + 16 * SROW;
    float* RED  = PART + 8 * 16 * 64;
    float* RMAX = RED + 256;
    float* RINV = RMAX + 16;

    const int tid     = threadIdx.x;
    const int w       = tid >> 5;          // wave id 0..7
    const int lane    = tid & 31;
    const int lm      = lane & 15;         // M (A) / N (B,C) index within tile
    const int hk      = (lane >> 4) * 2;   // K sub-offset for A/B fragments
    const int mbase   = (lane >> 4) * 8;   // C-tile M base for this half-wave
    const int bIdx    = blockIdx.x >> 7;   // 128 row-blocks per batch
    const int rowBase = (blockIdx.x & 127) * 16;

    // ---------------- Phase 1: S = scale * Q K^T + bias (into LDS) ----------
    // Hoist Q A-fragments: A is 16x4 fp32; lane holds (m=lm, K=hk..hk+1).
    const v2f* qp = (const v2f*)(q + ((size_t)bIdx * T_ + rowBase + lm) * D_ + hk);
    v2f aq[16];
#pragma unroll
    for (int kk = 0; kk < 16; ++kk) aq[kk] = qp[kk * 2];

    for (int nt = w; nt < T_ / 16; nt += 8) {
        const int keyBase = nt * 16;
        const v2f* kp = (const v2f*)(k + ((size_t)bIdx * T_ + keyBase + lm) * D_ + hk);
        v8f acc = {};
#pragma unroll
        for (int kk = 0; kk < 16; ++kk) {
            v2f bk = kp[kk * 2];   // B[kdim][n] = K[keyBase+n][kk*4+kdim]
            acc = __builtin_amdgcn_wmma_f32_16x16x4_f32(
                false, aq[kk], false, bk, (short)0, acc, false, false);
        }
        // C layout: vgpr r -> M = mbase + r, N = lm
        const float* bp = bias + ((size_t)bIdx * T_ + rowBase + mbase) * T_ + keyBase + lm;
#pragma unroll
        for (int r = 0; r < 8; ++r) {
            S[(mbase + r) * SROW + keyBase + lm] = acc[r] * SCALE + bp[(size_t)r * T_];
        }
    }
    __syncthreads();

    // ---------------- Phase 2: softmax over keys (in LDS) -------------------
    const int row = tid >> 4;       // 0..15
    const int seg = tid & 15;       // 16 segments per row, cols strided by 16
    float* Sr = S + row * SROW;

    float mx = -3.4e38f;
#pragma unroll 8
    for (int j = 0; j < 128; ++j) mx = fmaxf(mx, Sr[seg + j * 16]);
    RED[tid] = mx;
    __syncthreads();
    if (tid < 16) {
        float m2 = RED[tid * 16];
#pragma unroll
        for (int i = 1; i < 16; ++i) m2 = fmaxf(m2, RED[tid * 16 + i]);
        RMAX[tid] = m2;
    }
    __syncthreads();

    const float rm = RMAX[row];
    float sum = 0.0f;
#pragma unroll 8
    for (int j = 0; j < 128; ++j) {
        const int c = seg + j * 16;
        float e = __expf(Sr[c] - rm);
        Sr[c] = e;                  // keep unnormalized exp for PV
        sum += e;
    }
    RED[tid] = sum;
    __syncthreads();
    if (tid < 16) {
        float s2 = 0.0f;
#pragma unroll
        for (int i = 0; i < 16; ++i) s2 += RED[tid * 16 + i];
        RINV[tid] = 1.0f / s2;
    }
    __syncthreads();

    // Normalized attention writeout: float4-flattened mapping.
    // 32768 floats = 8192 float4 over [16 rows][512 float4-cols].
    // Consecutive lanes -> consecutive 16B chunks: coalesced b128 stores and
    // conflict-free ds_load_b128 (SROW divisible by 4).
    {
        float* ap = attn + ((size_t)bIdx * T_ + rowBase) * T_;
#pragma unroll 4
        for (int i = 0; i < 32; ++i) {
            const int f  = tid + i * 256;       // float4 index 0..8191
            const int rr = f >> 9;              // row 0..15
            const int cc = (f & 511) * 4;       // col 0..2044
            v4f p = *(const v4f*)(S + rr * SROW + cc);
            const float ri = RINV[rr];
            p *= ri;
            *(v4f*)(ap + (size_t)rr * T_ + cc) = p;
        }
    }

    // ---------------- Phase 3: out = P V (WMMA, per-wave K chunks) ----------
    // Wave w handles keys [w*256, w*256+256). 4 accumulator tiles cover D=64.
    v8f oacc[4];
#pragma unroll
    for (int nt = 0; nt < 4; ++nt) oacc[nt] = (v8f){};

    const int kc = w * 256;
    const float* vb = v + (size_t)bIdx * T_ * D_;
    for (int kk = 0; kk < 64; ++kk) {
        const int key0 = kc + kk * 4;
        // A frag from LDS: P[lm][key0+hk .. +1]  (ds_load_b64, conflict-free)
        v2f pa = *(const v2f*)(S + lm * SROW + key0 + hk);
        // B frags from V: B[kdim][n] = V[key0+kdim][nt*16+n]
        const float* vr = vb + (size_t)(key0 + hk) * D_ + lm;
#pragma unroll
        for (int nt = 0; nt < 4; ++nt) {
            v2f bv;
            bv.x = vr[nt * 16];
            bv.y = vr[nt * 16 + D_];
            oacc[nt] = __builtin_amdgcn_wmma_f32_16x16x4_f32(
                false, pa, false, bv, (short)0, oacc[nt], false, false);
        }
    }

    // Stash per-wave partials, reduce across the 8 waves, scale by 1/rowsum.
    {
        float* part = PART + w * (16 * 64);
#pragma unroll
        for (int nt = 0; nt < 4; ++nt)
#pragma unroll
            for (int r = 0; r < 8; ++r)
                part[(mbase + r) * 64 + nt * 16 + lm] = oacc[nt][r];
    }
    __syncthreads();

    // Final reduce + write: 16x64 = 256 float4; one per thread.
    {
        float* op = out + ((size_t)bIdx * T_ + rowBase) * D_;
        const int m  = tid >> 4;            // row 0..15
        const int dc = (tid & 15) * 4;      // col 0..60
        v4f sacc = {};
#pragma unroll
        for (int ww = 0; ww < 8; ++ww)
            sacc += *(const v4f*)(PART + ww * 1024 + m * 64 + dc);
        sacc *= RINV[m];
        *(v4f*)(op + (size_t)m * D_ + dc) = sacc;
    }
}

extern "C" void kernel_launch(void* const* d_in, const int* in_sizes, int n_in,
                              void* d_out, int out_size, void* d_ws, size_t ws_size,
                              hipStream_t stream) {
    (void)in_sizes; (void)n_in; (void)d_ws; (void)ws_size; (void)out_size;
    const float* q    = (const float*)d_in[0];
    const float* k    = (const float*)d_in[1];
    const float* v    = (const float*)d_in[2];
    const float* bias = (const float*)d_in[3];
    float* out  = (float*)d_out;                         // [B,T,D]
    float* attn = out + (size_t)B_ * T_ * D_;            // [B,T,T]

    const int grid = B_ * (T_ / 16);                     // 2048 workgroups
    sdpa_fused_kernel<<<grid, 256, SMEM_BYTES, stream>>>(q, k, v, bias, out, attn);
}